// HierarchicalRouter_83897891160583
// MI455X (gfx1250) — compile-verified
//
#include <hip/hip_runtime.h>

#define B_TOK 32768
#define H_DIM 4096
#define G_NUM 16
#define EPG_N 16
#define E_NUM 256
#define WPB   8                    // waves per block (256 threads, wave32)
#define TILES (B_TOK / 16)         // 2048 16-token tiles

typedef __attribute__((ext_vector_type(2))) float v2f;
typedef __attribute__((ext_vector_type(4))) float v4f;
typedef __attribute__((ext_vector_type(8))) float v8f;

static __device__ __forceinline__ v8f wmma4(v2f a, v2f b, v8f c) {
  // D = A(16x4 f32) * B(4x16 f32) + C(16x16 f32), fp32 tensor path
  return __builtin_amdgcn_wmma_f32_16x16x4_f32(false, a, false, b, (short)0, c,
                                               false, false);
}

// 16x16 tile GEMM over K=H_DIM with software-pipelined b128 feeds.
//   ap = hidden row(lane%16) + 4*(lane/16) floats   (streamed -> NT loads)
//   bp = weight row(lane%16) + 4*(lane/16) floats   (L2-resident -> RT loads)
// One b128/lane covers 8 K-columns = 2 WMMA steps (same K-permutation on A & B).
// 8 accumulators: each updated exactly once per pipeline stage so the register
// allocator can pin the 8-VGPR tuples (no accumulator copy ping-pong).
static __device__ __forceinline__ v8f tile_gemm(const float* __restrict__ ap,
                                                const float* __restrict__ bp) {
  constexpr int STEPS = H_DIM / 8;  // 512
  constexpr int PF = 4;             // prefetch depth: 4 a/b pairs in flight
  v8f c0 = {}; v8f c1 = {}; v8f c2 = {}; v8f c3 = {};
  v8f c4 = {}; v8f c5 = {}; v8f c6 = {}; v8f c7 = {};
  v4f a[PF], b[PF];
#pragma unroll
  for (int i = 0; i < PF; ++i) {
    a[i] = __builtin_nontemporal_load((const v4f*)(ap + 8 * i));
    b[i] = *(const v4f*)(bp + 8 * i);
  }
  for (int s = PF; s < STEPS; s += PF) {
    v4f ca0 = a[0], cb0 = b[0], ca1 = a[1], cb1 = b[1];
    v4f ca2 = a[2], cb2 = b[2], ca3 = a[3], cb3 = b[3];
#pragma unroll
    for (int i = 0; i < PF; ++i) {
      a[i] = __builtin_nontemporal_load((const v4f*)(ap + 8 * (s + i)));
      b[i] = *(const v4f*)(bp + 8 * (s + i));
    }
    c0 = wmma4(ca0.lo, cb0.lo, c0); c1 = wmma4(ca0.hi, cb0.hi, c1);
    c2 = wmma4(ca1.lo, cb1.lo, c2); c3 = wmma4(ca1.hi, cb1.hi, c3);
    c4 = wmma4(ca2.lo, cb2.lo, c4); c5 = wmma4(ca2.hi, cb2.hi, c5);
    c6 = wmma4(ca3.lo, cb3.lo, c6); c7 = wmma4(ca3.hi, cb3.hi, c7);
  }
  c0 = wmma4(a[0].lo, b[0].lo, c0); c1 = wmma4(a[0].hi, b[0].hi, c1);
  c2 = wmma4(a[1].lo, b[1].lo, c2); c3 = wmma4(a[1].hi, b[1].hi, c3);
  c4 = wmma4(a[2].lo, b[2].lo, c4); c5 = wmma4(a[2].hi, b[2].hi, c5);
  c6 = wmma4(a[3].lo, b[3].lo, c6); c7 = wmma4(a[3].hi, b[3].hi, c7);
  return ((c0 + c1) + (c2 + c3)) + ((c4 + c5) + (c6 + c7));
}

__global__ void hr_init_counts(int* __restrict__ g_count) {
  if (threadIdx.x < G_NUM) g_count[threadIdx.x] = 0;
}

// Phase 1: group logits via WMMA, per-token argmax over 16 groups, bucket append.
__global__ __launch_bounds__(256) void hr_group_route(
    const float* __restrict__ hidden, const float* __restrict__ group_w,
    int* __restrict__ g_count, int* __restrict__ g_list) {
  const int lane = threadIdx.x & 31;
  const int wave = threadIdx.x >> 5;
  const int tile = blockIdx.x * WPB + wave;
  const int tok0 = tile * 16;
  const int m = lane & 15;   // A-row / B-col owned by this lane
  const int h = lane >> 4;   // wave half -> K slots {2h, 2h+1}

  const float* ap = hidden  + (size_t)(tok0 + m) * H_DIM + 4 * h;
  const float* bp = group_w + (size_t)m * H_DIM + 4 * h;   // G_NUM==16 rows

  v8f c = tile_gemm(ap, bp);  // c[i] = logits[token tok0+i+8h][group m]

  // Branchless cross-lane argmax over 16 groups (cols live in 16 lanes per half).
  float val[8]; int idx[8];
#pragma unroll
  for (int i = 0; i < 8; ++i) { val[i] = c[i]; idx[i] = m; }
#pragma unroll
  for (int off = 1; off < 16; off <<= 1) {
#pragma unroll
    for (int i = 0; i < 8; ++i) {
      const float ov = __shfl_xor(val[i], off, 32);
      const int   oi = __shfl_xor(idx[i], off, 32);
      const int take = (int)(ov > val[i]) | ((int)(ov == val[i]) & (int)(oi < idx[i]));
      val[i] = take ? ov : val[i];
      idx[i] = take ? oi : idx[i];
    }
  }
  // lanes 0 and 16 hold argmax for rows 0..7 / 8..15; append to buckets
  if (m == 0) {
    const int rbase = tok0 + 8 * h;
#pragma unroll
    for (int i = 0; i < 8; ++i) {
      const int g = idx[i];
      const int pos = atomicAdd(&g_count[g], 1);
      g_list[g * B_TOK + pos] = rbase + i;
    }
  }
}

// Phase 2: per group, gather 16 bucketed tokens, WMMA vs that group's 16 experts,
// fused softmax/argmax, write full output rows (zeros + scattered logits).
__global__ __launch_bounds__(256) void hr_expert_route(
    const float* __restrict__ hidden, const float* __restrict__ expert_w,
    const int* __restrict__ g_count, const int* __restrict__ g_list,
    float* __restrict__ out_logits, float* __restrict__ out_sel,
    float* __restrict__ out_wt) {
  const int lane = threadIdx.x & 31;
  const int wave = threadIdx.x >> 5;
  const int g = blockIdx.y;
  const int cnt = g_count[g];
  const int start = (blockIdx.x * WPB + wave) * 16;
  if (start >= cnt) return;                       // wave-uniform: EXEC stays all-1

  const int m = lane & 15;
  const int h = lane >> 4;
  const int slot = min(start + m, cnt - 1);       // clamp ragged tail (dup rows)
  const int my_tok = g_list[g * B_TOK + slot];

  const float* ap = hidden + (size_t)my_tok * H_DIM + 4 * h;
  const float* bp = expert_w + ((size_t)g * EPG_N + m) * H_DIM + 4 * h;

  v8f c = tile_gemm(ap, bp);  // c[i] = logit[row i+8h][expert m] of group g

  // Token index of the 8 rows this half-wave tracks (full-EXEC shuffles first)
  int tokr[8];
#pragma unroll
  for (int i = 0; i < 8; ++i) tokr[i] = __shfl(my_tok, i + 8 * h, 32);

  // Branchless argmax over 16 experts per row
  float lg[8]; float val[8]; int idx[8];
#pragma unroll
  for (int i = 0; i < 8; ++i) { lg[i] = c[i]; val[i] = c[i]; idx[i] = m; }
#pragma unroll
  for (int off = 1; off < 16; off <<= 1) {
#pragma unroll
    for (int i = 0; i < 8; ++i) {
      const float ov = __shfl_xor(val[i], off, 32);
      const int   oi = __shfl_xor(idx[i], off, 32);
      const int take = (int)(ov > val[i]) | ((int)(ov == val[i]) & (int)(oi < idx[i]));
      val[i] = take ? ov : val[i];
      idx[i] = take ? oi : idx[i];
    }
  }
  // softmax denominator: sum over 16 experts of exp(l - rowmax)
  float ex[8];
#pragma unroll
  for (int i = 0; i < 8; ++i) ex[i] = __expf(lg[i] - val[i]);
#pragma unroll
  for (int off = 1; off < 16; off <<= 1)
#pragma unroll
    for (int i = 0; i < 8; ++i) ex[i] += __shfl_xor(ex[i], off, 32);

  // Zero-fill output rows: half-wave h owns rows 8h..8h+7 (tokens in tokr[]);
  // 16 lanes x 16 floats (4x b128 NT stores) cover each 256-wide row.
#pragma unroll
  for (int i = 0; i < 8; ++i) {
    const int r = i + 8 * h;
    if (start + r < cnt) {
      float* rowp = out_logits + (size_t)tokr[i] * E_NUM + m * 16;
      const v4f z = {};
      __builtin_nontemporal_store(z, (v4f*)(rowp));
      __builtin_nontemporal_store(z, (v4f*)(rowp + 4));
      __builtin_nontemporal_store(z, (v4f*)(rowp + 8));
      __builtin_nontemporal_store(z, (v4f*)(rowp + 12));
    }
  }
  // Scatter this lane's expert logit into column g*16+m of its 8 rows
  // (same-wave stores land in program order after the zero-fill).
#pragma unroll
  for (int i = 0; i < 8; ++i) {
    const int r = i + 8 * h;
    if (start + r < cnt)
      out_logits[(size_t)tokr[i] * E_NUM + g * EPG_N + m] = lg[i];
  }
  // Selected expert (global id) and its softmax weight = 1/sum(exp(l-max))
  if (m == 0) {
#pragma unroll
    for (int i = 0; i < 8; ++i) {
      const int r = i + 8 * h;
      if (start + r < cnt) {
        out_sel[tokr[i]] = (float)(g * EPG_N + idx[i]);
        out_wt[tokr[i]]  = 1.0f / ex[i];
      }
    }
  }
}

extern "C" void kernel_launch(void* const* d_in, const int* in_sizes, int n_in,
                              void* d_out, int out_size, void* d_ws, size_t ws_size,
                              hipStream_t stream) {
  const float* hidden   = (const float*)d_in[0];  // [B, H]
  const float* group_w  = (const float*)d_in[1];  // [G, H]
  const float* expert_w = (const float*)d_in[2];  // [G, EPG, H]
  float* out = (float*)d_out;
  float* out_logits = out;                                   // [B, E]
  float* out_sel    = out + (size_t)B_TOK * E_NUM;           // [B]
  float* out_wt     = out + (size_t)B_TOK * E_NUM + B_TOK;   // [B]

  int* g_count = (int*)d_ws;         // 16 counters (padded to 64 ints)
  int* g_list  = (int*)d_ws + 64;    // [G][B] token buckets (2 MB)

  hr_init_counts<<<dim3(1), dim3(32), 0, stream>>>(g_count);
  hr_group_route<<<dim3(TILES / WPB), dim3(256), 0, stream>>>(
      hidden, group_w, g_count, g_list);
  hr_expert_route<<<dim3(TILES / WPB, G_NUM), dim3(256), 0, stream>>>(
      hidden, expert_w, g_count, g_list, out_logits, out_sel, out_wt);
}